// GISMo_51771535786132
// MI455X (gfx1250) — compile-verified
//
#include <hip/hip_runtime.h>
#include <hip/hip_bf16.h>

#define N_NODES  50000
#define N_EDGES  800000
#define D        128
#define VOCAB    30000
#define BATCH    1024
#define CTX_LEN  50
#define BN_EPS   1e-5f
#define WPITCH   132   // 128 + 4 dword pad -> 16 distinct LDS banks for row-strided frags

typedef __attribute__((ext_vector_type(2))) float v2f;
typedef __attribute__((ext_vector_type(8))) float v8f;
typedef __attribute__((ext_vector_type(4))) unsigned int u32x4;
typedef __attribute__((ext_vector_type(4))) int i32x4;
typedef __attribute__((ext_vector_type(8))) int i32x8;

#if defined(__has_builtin)
# if __has_builtin(__builtin_amdgcn_tensor_load_to_lds)
#  define HAVE_TDM 1
# endif
#endif

// ---------------------------------------------------------------------------
// Guaranteed-hardware f32 atomic add (no-return form -> STOREcnt; s_endpgm
// performs an implicit wait-idle, so no explicit wait needed).
// ---------------------------------------------------------------------------
__device__ __forceinline__ void atomic_add4_f32(float* dst, float a, float b,
                                                float c, float d) {
    asm volatile("global_atomic_add_f32 %0, %1, off\n\t"
                 "global_atomic_add_f32 %0, %2, off offset:4\n\t"
                 "global_atomic_add_f32 %0, %3, off offset:8\n\t"
                 "global_atomic_add_f32 %0, %4, off offset:12"
                 :: "v"(dst), "v"(a), "v"(b), "v"(c), "v"(d)
                 : "memory");
}

// ---------------------------------------------------------------------------
// Edge-parallel SpMM: agg[row[e], :] += norm[e] * h[col[e], :]
// One wave per edge; lane handles 4 consecutive floats (L2-resident data).
// ---------------------------------------------------------------------------
__global__ void spmm_edges(const int* __restrict__ row, const int* __restrict__ col,
                           const float* __restrict__ norm,
                           const float* __restrict__ h, float* __restrict__ agg,
                           int nEdges) {
    int wave = (blockIdx.x * blockDim.x + threadIdx.x) >> 5;
    int lane = threadIdx.x & 31;
    if (wave >= nEdges) return;
    int r = row[wave];
    int c = col[wave];
    float w = norm[wave];
    const float4* src = (const float4*)(h + (size_t)c * D);
    float4 v = src[lane];
    float* dst = agg + (size_t)r * D + lane * 4;
    atomic_add4_f32(dst, v.x * w, v.y * w, v.z * w, v.w * w);
}

// ---------------------------------------------------------------------------
// GCN layer GEMM: z = relu(agg @ W^T), W is [128,128].
// Block = 256 threads = 8 waves = 8 column tiles of one 16-row node tile.
// W staged into LDS (pitch 132) by the Tensor Data Mover (async DMA,
// TENSORcnt); A tile staged cooperatively. WMMA fed entirely from LDS.
// ---------------------------------------------------------------------------
__global__ void __launch_bounds__(256)
gcn_gemm_wmma(const float* __restrict__ X,   // agg [N_NODES, 128]
              const float* __restrict__ W,   // [128, 128] row-major
              float* __restrict__ Y) {       // z   [N_NODES, 128]
    __shared__ float smem[128 * WPITCH + 16 * WPITCH];  // 74.25 KB
    float* Wl = smem;                  // LDS offset 0 (single static allocation)
    float* Al = smem + 128 * WPITCH;

    int t = threadIdx.x;
    int mBase = blockIdx.x * 16;

#ifdef HAVE_TDM
    if ((t >> 5) == 0) {
        // ---- Tensor DMA descriptor (D#): 128x128 f32, pad 4dw per 128dw ----
        u32x4 g0; i32x8 g1; i32x4 g2 = {0, 0, 0, 0}; i32x4 g3 = {0, 0, 0, 0};
        unsigned long long ga = (unsigned long long)(size_t)W;
        g0[0] = 1u;                                    // count=1, user mode
        g0[1] = 0u;                                    // lds_addr = 0 (Wl)
        g0[2] = (unsigned)(ga & 0xffffffffu);          // global_addr[31:0]
        g0[3] = (unsigned)((ga >> 32) & 0x01ffffffu)   // global_addr[56:32]
              | (2u << 30);                            // type = 2 ("image")
        g1[0] = (2 << 16)      // data_size = 4 bytes
              | (1 << 20)      // pad_enable
              | (6 << 22)      // pad_interval: 2^(6+1)=128 dwords
              | (3 << 25);     // pad_amount: 3+1 = 4 dwords
        g1[1] = (128 << 16);   // tensor_dim0 = 128
        g1[2] = (128 << 16);   // tensor_dim1 = 128
        g1[3] = (128 << 16);   // tile_dim0   = 128
        g1[4] = 128;           // tile_dim1   = 128 (tile_dim2 = 0)
        g1[5] = 128;           // tensor_dim0_stride = 128
        g1[6] = 0;
        g1[7] = 0;
#if __clang_major__ >= 23
        i32x8 g4 = {0, 0, 0, 0, 0, 0, 0, 0};
        __builtin_amdgcn_tensor_load_to_lds(g0, g1, g2, g3, g4, 0);
#else
        __builtin_amdgcn_tensor_load_to_lds(g0, g1, g2, g3, 0);
#endif
    }
#else
    // Fallback: cooperative global -> LDS copy of W with padded pitch
    for (int idx = t; idx < 128 * 32; idx += 256) {    // float4 units
        int rw = idx >> 5, cf = idx & 31;
        float4 v = ((const float4*)(W + (size_t)rw * 128))[cf];
        float* dp = Wl + rw * WPITCH + cf * 4;
        dp[0] = v.x; dp[1] = v.y; dp[2] = v.z; dp[3] = v.w;
    }
#endif

    // ---- stage A tile (16 x 128) into LDS, pitch 132 ----
#pragma unroll
    for (int i = 0; i < 2; i++) {
        int idx = t + i * 256;                         // 512 float4 units total
        int rw = idx >> 5, cf = idx & 31;
        float4 v = ((const float4*)(X + (size_t)(mBase + rw) * D))[cf];
        float* dp = Al + rw * WPITCH + cf * 4;
        dp[0] = v.x; dp[1] = v.y; dp[2] = v.z; dp[3] = v.w;
    }

#ifdef HAVE_TDM
    if ((t >> 5) == 0) __builtin_amdgcn_s_wait_tensorcnt(0);
#endif
    __syncthreads();

    int wave = t >> 5;                 // column tile 0..7
    int lane = t & 31;
    int half = lane >> 4, lm = lane & 15;
    const float* ar = Al + lm * WPITCH + 2 * half;
    const float* br = Wl + (wave * 16 + lm) * WPITCH + 2 * half;

    v8f c = {};
#pragma unroll
    for (int k = 0; k < D; k += 4) {
        v2f a, b;
        a.x = ar[k]; a.y = ar[k + 1];
        b.x = br[k]; b.y = br[k + 1];
        c = __builtin_amdgcn_wmma_f32_16x16x4_f32(false, a, false, b,
                                                  (short)0, c, false, false);
    }

#pragma unroll
    for (int r = 0; r < 8; r++) {
        int mm = mBase + r + 8 * half;
        int nn = wave * 16 + lm;
        Y[(size_t)mm * D + nn] = fmaxf(c[r], 0.0f);
    }
}

// ---------------------------------------------------------------------------
// Generic f32 WMMA GEMM for the small fusion/projection layers.
// ---------------------------------------------------------------------------
__global__ void gemm_wmma_f32(const float* __restrict__ X, int ldx,
                              const float* __restrict__ W,      // [N,K] row-major
                              float* __restrict__ Y, int ldy,
                              const float* __restrict__ bias,
                              int M, int N, int K, int relu) {
    int waveId = (blockIdx.x * blockDim.x + threadIdx.x) >> 5;
    int lane   = threadIdx.x & 31;
    int nTiles = N >> 4;
    int mTile  = waveId / nTiles;
    int nTile  = waveId - mTile * nTiles;
    if (mTile * 16 >= M) return;          // wave-uniform exit, EXEC stays full
    int mBase = mTile << 4;
    int nBase = nTile << 4;
    int half  = lane >> 4;
    int lm    = lane & 15;

    const float* xr = X + (size_t)(mBase + lm) * ldx + 2 * half;
    const float* wr = W + (size_t)(nBase + lm) * K + 2 * half;

    v8f c = {};
    for (int k = 0; k < K; k += 4) {
        v2f a, b;
        a.x = xr[k];  a.y = xr[k + 1];
        b.x = wr[k];  b.y = wr[k + 1];
        c = __builtin_amdgcn_wmma_f32_16x16x4_f32(false, a, false, b,
                                                  (short)0, c, false, false);
    }

#pragma unroll
    for (int r = 0; r < 8; r++) {
        int mm = mBase + r + 8 * half;
        int nn = nBase + lm;
        float v = c[r];
        if (bias) v += bias[nn];
        if (relu) v = fmaxf(v, 0.0f);
        Y[(size_t)mm * ldy + nn] = v;
    }
}

// ---------------------------------------------------------------------------
// BatchNorm statistics: one block per feature d (deterministic reduction).
// ---------------------------------------------------------------------------
__global__ void bn_stats(const float* __restrict__ z, float* __restrict__ mu,
                         float* __restrict__ var, int N) {
    int d = blockIdx.x;
    int t = threadIdx.x;
    float s = 0.f, s2 = 0.f;
    for (int n = t; n < N; n += blockDim.x) {
        float v = z[(size_t)n * D + d];
        s += v; s2 += v * v;
    }
    __shared__ float sh[256], sh2[256];
    sh[t] = s; sh2[t] = s2;
    __syncthreads();
    for (int off = 128; off > 0; off >>= 1) {
        if (t < off) { sh[t] += sh[t + off]; sh2[t] += sh2[t + off]; }
        __syncthreads();
    }
    if (t == 0) {
        float m = sh[0] / (float)N;
        mu[d]  = m;
        var[d] = sh2[0] / (float)N - m * m;   // biased var, matches jnp.var
    }
}

// h_out = h_in + (z - mu) * rsqrt(var+eps) * gamma + beta
__global__ void bn_apply(const float* __restrict__ hin, const float* __restrict__ z,
                         const float* __restrict__ mu, const float* __restrict__ var,
                         const float* __restrict__ gamma, const float* __restrict__ beta,
                         float* __restrict__ hout, int total) {
    int i = blockIdx.x * blockDim.x + threadIdx.x;
    if (i >= total) return;
    int d = i & (D - 1);
    float zn = (z[i] - mu[d]) * rsqrtf(var[d] + BN_EPS);
    hout[i] = hin[i] + zn * gamma[d] + beta[d];
}

// ---------------------------------------------------------------------------
// Context attention pooling: one block (128 threads = 4 waves) per batch row.
// Writes ctx_emb directly into q_in[:, 0:128] (row stride 2D).
// ---------------------------------------------------------------------------
__global__ void attn_pool(const int* __restrict__ ctx_ids,
                          const float* __restrict__ emb,
                          const float* __restrict__ attn_w,
                          const float* __restrict__ attn_b,
                          float* __restrict__ qin) {
    int b = blockIdx.x;
    int t = threadIdx.x;           // 0..127
    __shared__ float logits[CTX_LEN];
    __shared__ int   ids[CTX_LEN];
    __shared__ float invSum;
    if (t < CTX_LEN) ids[t] = ctx_ids[b * CTX_LEN + t];
    __syncthreads();

    int wave = t >> 5, lane = t & 31;
    for (int l = wave; l < CTX_LEN; l += 4) {
        const float4* cp = (const float4*)(emb + (size_t)ids[l] * D);
        const float4* wp = (const float4*)attn_w;
        float4 cv = cp[lane];
        float4 wv = wp[lane];
        float p = cv.x * wv.x + cv.y * wv.y + cv.z * wv.z + cv.w * wv.w;
        for (int off = 16; off > 0; off >>= 1) p += __shfl_xor(p, off, 32);
        if (lane == 0) logits[l] = p + attn_b[0];
    }
    __syncthreads();
    if (t == 0) {
        float m = -1e30f;
        for (int l = 0; l < CTX_LEN; l++) m = fmaxf(m, logits[l]);
        float s = 0.f;
        for (int l = 0; l < CTX_LEN; l++) { float e = __expf(logits[l] - m); logits[l] = e; s += e; }
        invSum = 1.0f / s;
    }
    __syncthreads();
    float inv = invSum;
    float acc = 0.f;
    for (int l = 0; l < CTX_LEN; l++)
        acc += logits[l] * inv * emb[(size_t)ids[l] * D + t];
    qin[(size_t)b * (2 * D) + t] = acc;
}

// Build fusion input: X[b, 0:128] = emb[miss], X[b, 128:256] = masked graph gather
__global__ void build_fuse(const int* __restrict__ miss_ids,
                           const int* __restrict__ v2fg,
                           const float* __restrict__ emb,
                           const float* __restrict__ gemb,
                           float* __restrict__ X) {
    int b = blockIdx.x, d = threadIdx.x;   // 128 threads
    int mid = miss_ids[b];
    X[(size_t)b * (2 * D) + d] = emb[(size_t)mid * D + d];
    int fg = v2fg[mid];
    X[(size_t)b * (2 * D) + D + d] = (fg >= 0) ? gemb[(size_t)fg * D + d] : 0.0f;
}

// ---------------------------------------------------------------------------
static inline void launch_gemm(const float* X, int ldx, const float* W,
                               float* Y, int ldy, const float* bias,
                               int M, int N, int K, int relu, hipStream_t s) {
    int tiles = (M / 16) * (N / 16);
    int blocks = (tiles + 7) / 8;
    gemm_wmma_f32<<<blocks, 256, 0, s>>>(X, ldx, W, Y, ldy, bias, M, N, K, relu);
}

extern "C" void kernel_launch(void* const* d_in, const int* in_sizes, int n_in,
                              void* d_out, int out_size, void* d_ws, size_t ws_size,
                              hipStream_t stream) {
    const int*   edge_index  = (const int*)d_in[0];     // [2, E]
    const float* norm        = (const float*)d_in[1];   // [E]
    const int*   ctx_ids     = (const int*)d_in[2];     // [B, L]
    const int*   miss_ids    = (const int*)d_in[3];     // [B]
    const int*   vocab_to_fg = (const int*)d_in[4];     // [VOCAB]
    const float* emb_table   = (const float*)d_in[5];   // [VOCAB, D]
    const float* fg_emb      = (const float*)d_in[6];   // [N, D]
    const float* gc_w        = (const float*)d_in[7];   // [2, D, D]
    const float* bn_gamma    = (const float*)d_in[8];   // [2, D]
    const float* bn_beta     = (const float*)d_in[9];   // [2, D]
    const float* attn_w      = (const float*)d_in[10];  // [D]
    const float* attn_b      = (const float*)d_in[11];  // [1]
    const float* fusion_w    = (const float*)d_in[12];  // [D, 2D]
    const float* fusion_b    = (const float*)d_in[13];  // [D]
    const float* proj1_w     = (const float*)d_in[14];  // [2D, 2D]
    const float* proj1_b     = (const float*)d_in[15];  // [2D]
    const float* proj2_w     = (const float*)d_in[16];  // [D, 2D]
    const float* proj2_b     = (const float*)d_in[17];  // [D]

    const int E = in_sizes[0] / 2;
    const int* e_row = edge_index;
    const int* e_col = edge_index + E;

    // d_out layout: query [B, D] first, then graph_embs [N, D]
    float* query_out = (float*)d_out;
    float* graph_out = (float*)d_out + (size_t)BATCH * D;

    size_t off = 0;
    auto carve = [&](size_t bytes) -> float* {
        off = (off + 255) & ~(size_t)255;
        float* p = (float*)((char*)d_ws + off);
        off += bytes;
        return p;
    };
    const size_t ND = (size_t)N_NODES * D * sizeof(float);
    float* h_ws   = carve(ND);
    float* agg    = carve(ND);
    float* z      = carve(ND);
    float* mu     = carve(D * sizeof(float));
    float* var    = carve(D * sizeof(float));
    float* X_fuse = carve((size_t)BATCH * 2 * D * sizeof(float));
    float* q_in   = carve((size_t)BATCH * 2 * D * sizeof(float));
    float* q1     = carve((size_t)BATCH * 2 * D * sizeof(float));
    (void)ws_size;

    const int totalND = N_NODES * D;
    const int spmmBlocks = (E * 32 + 255) / 256;

    // ---------------- GCN layer 0: reads fg_emb, writes h_ws ----------------
    hipMemsetAsync(agg, 0, ND, stream);
    spmm_edges<<<spmmBlocks, 256, 0, stream>>>(e_row, e_col, norm, fg_emb, agg, E);
    gcn_gemm_wmma<<<N_NODES / 16, 256, 0, stream>>>(agg, gc_w + 0 * D * D, z);
    bn_stats<<<D, 256, 0, stream>>>(z, mu, var, N_NODES);
    bn_apply<<<(totalND + 255) / 256, 256, 0, stream>>>(fg_emb, z, mu, var,
             bn_gamma + 0 * D, bn_beta + 0 * D, h_ws, totalND);

    // ------------- GCN layer 1: reads h_ws, writes graph_out ---------------
    hipMemsetAsync(agg, 0, ND, stream);
    spmm_edges<<<spmmBlocks, 256, 0, stream>>>(e_row, e_col, norm, h_ws, agg, E);
    gcn_gemm_wmma<<<N_NODES / 16, 256, 0, stream>>>(agg, gc_w + 1 * D * D, z);
    bn_stats<<<D, 256, 0, stream>>>(z, mu, var, N_NODES);
    bn_apply<<<(totalND + 255) / 256, 256, 0, stream>>>(h_ws, z, mu, var,
             bn_gamma + 1 * D, bn_beta + 1 * D, graph_out, totalND);

    // ---------------- context attention pooling -> q_in[:, 0:128] ----------
    attn_pool<<<BATCH, D, 0, stream>>>(ctx_ids, emb_table, attn_w, attn_b, q_in);

    // ---------------- fusion: X_fuse -> q_in[:, 128:256] -------------------
    build_fuse<<<BATCH, D, 0, stream>>>(miss_ids, vocab_to_fg, emb_table, graph_out, X_fuse);
    launch_gemm(X_fuse, 2 * D, fusion_w, q_in + D, 2 * D, fusion_b,
                BATCH, D, 2 * D, 0, stream);

    // ---------------- projection MLP ---------------------------------------
    launch_gemm(q_in, 2 * D, proj1_w, q1, 2 * D, proj1_b, BATCH, 2 * D, 2 * D, 1, stream);
    launch_gemm(q1, 2 * D, proj2_w, query_out, D, proj2_b, BATCH, D, 2 * D, 0, stream);
}